// direction_Loss_83210696392941
// MI455X (gfx1250) — compile-verified
//
#include <hip/hip_runtime.h>

// CDNA5 / gfx1250: wave32, WMMA 16x16x4 f32.
typedef __attribute__((ext_vector_type(2))) float v2f;
typedef __attribute__((ext_vector_type(8))) float v8f;

// Wave32-wide sum via V_WMMA_F32_16X16X4_F32 with B = ones.
// A layout (32-bit A 16x4): lanes 0-15 -> {K0,K1}, lanes 16-31 -> {K2,K3}, M = lane%16.
// With a = {acc, 0}: D[m][n] = acc(lane m) + acc(lane m+16) for every n.
// C/D layout: VGPR v, lanes 0-15 -> M=v, lanes 16-31 -> M=v+8. So
// sum_{v} c[v]  (per lane)  +  shfl_xor(.,16)  = sum over all 32 lanes. Exact f32 adds.
__device__ __forceinline__ float wave_sum_wmma(float acc) {
  v2f a; a[0] = acc;  a[1] = 0.0f;
  v2f b; b[0] = 1.0f; b[1] = 1.0f;
  v8f c = {};
  // (neg_a, A, neg_b, B, c_mod, C, reuse_a, reuse_b)
  c = __builtin_amdgcn_wmma_f32_16x16x4_f32(false, a, false, b, (short)0, c,
                                            false, false);
  float s = ((c[0] + c[1]) + (c[2] + c[3])) + ((c[4] + c[5]) + (c[6] + c[7]));
  s += __shfl_xor(s, 16, 32);
  return s; // uniform across the wave
}

// Deterministic block(256)-wide sum: WMMA wave reduce -> 8 partials in LDS ->
// fixed-order add by thread 0. Valid in thread 0 only.
__device__ __forceinline__ float block_sum_256(float acc) {
  __shared__ float red[8];
  float s = wave_sum_wmma(acc);
  const int lane = threadIdx.x & 31;
  const int wave = threadIdx.x >> 5;
  if (lane == 0) red[wave] = s;
  __syncthreads();
  float t = 0.0f;
  if (threadIdx.x == 0) {
#pragma unroll
    for (int i = 0; i < 8; ++i) t += red[i];
  }
  return t;
}

// One block per row: e_dist_sq = sum_j (x0[r,j]-x1[r,j])^2, then per-row loss
// term. float4 -> global_load_b128 streaming loads (bandwidth-bound path).
__global__ __launch_bounds__(256) void row_eloss_kernel(
    const float* __restrict__ x0, const float* __restrict__ x1,
    const int* __restrict__ y, float* __restrict__ row_loss, int D) {
  const int row = blockIdx.x;
  const float4* __restrict__ p0 =
      reinterpret_cast<const float4*>(x0 + (size_t)row * (size_t)D);
  const float4* __restrict__ p1 =
      reinterpret_cast<const float4*>(x1 + (size_t)row * (size_t)D);
  const int nvec = D >> 2;

  float acc = 0.0f;
  for (int i = threadIdx.x; i < nvec; i += 256) {
    float4 a = p0[i];
    float4 b = p1[i];
    float dx = a.x - b.x, dy = a.y - b.y, dz = a.z - b.z, dw = a.w - b.w;
    acc = fmaf(dx, dx, acc);
    acc = fmaf(dy, dy, acc);
    acc = fmaf(dz, dz, acc);
    acc = fmaf(dw, dw, acc);
  }

  float e_dist_sq = block_sum_256(acc);
  if (threadIdx.x == 0) {
    float e_dist = sqrtf(e_dist_sq);
    float e_clamped = fmaxf(1.2f - e_dist, 0.0f);  // clamp(MARGIN - dist, 0)
    // y in {0,1}:  yf*e_dist_sq + (1-yf)*e_clamped
    row_loss[row] = (y[row] != 0) ? e_dist_sq : e_clamped;
  }
}

// Single block, fixed-stride, fixed-order final reduction -> loss scalar.
__global__ __launch_bounds__(256) void final_reduce_kernel(
    const float* __restrict__ row_loss, float* __restrict__ out, int N) {
  float acc = 0.0f;
  for (int i = threadIdx.x; i < N; i += 256) acc += row_loss[i];
  float total = block_sum_256(acc);
  if (threadIdx.x == 0) out[0] = total * (0.5f / (float)N);  // sum/2/N
}

extern "C" void kernel_launch(void* const* d_in, const int* in_sizes, int n_in,
                              void* d_out, int out_size, void* d_ws,
                              size_t ws_size, hipStream_t stream) {
  // setup_inputs order: x0, x1, x_c_0, x_c_1, y.  x_c_* are dead in the
  // reference's live output; only x0, x1, y feed the returned loss.
  const float* x0 = (const float*)d_in[0];
  const float* x1 = (const float*)d_in[1];
  const int* y = (const int*)d_in[4];
  const int N = in_sizes[4];          // 4096 rows
  const int D = in_sizes[0] / N;      // 4096 cols (divisible by 1024)

  float* row_loss = (float*)d_ws;     // N floats of scratch

  row_eloss_kernel<<<N, 256, 0, stream>>>(x0, x1, y, row_loss, D);
  final_reduce_kernel<<<1, 256, 0, stream>>>(row_loss, (float*)d_out, N);
}